// BayesianElementwiseLinear_23295902614120
// MI455X (gfx1250) — compile-verified
//
#include <hip/hip_runtime.h>

// ---------------------------------------------------------------------------
// BayesianElementwiseLinear for MI455X (gfx1250, wave32)
//   out[s,b,j] = x[b,j] * (mu[j] + softplus(rho[j]) * eps[s,j])
//   + two scalar log-prob reductions over the S*J weight samples.
// Bandwidth-bound: 64MB read + 256MB NT write -> ~14us floor @ 23.3 TB/s.
// ---------------------------------------------------------------------------

typedef float v2f __attribute__((ext_vector_type(2)));
typedef float v4f __attribute__((ext_vector_type(4)));
typedef float v8f __attribute__((ext_vector_type(8)));

#define JDIM 4096
#define BDIM 4096
#define SDIM 4
#define J4   (JDIM / 4)

// constants from the reference
#define LOG_SQRT_2PI 0.91893853320467274178f
#define LOG_HALF    -0.69314718055994530942f
#define LOG_SIGMA2  -6.21460809842219420495f   /* log(0.002) */
#define INV_2SIG2_2  125000.0f                 /* 1/(2*0.002^2) */

// Wave-level sum using V_WMMA_F32_16X16X4_F32 with A == ones:
//   D[m,n] = sum_k B[k,n]  (independent of B's lane<->(k,n) mapping,
//   every per-lane partial occupies exactly one B slot; the rest are 0).
// After the WMMA each lane holds one column-sum; 4 shfl_xor hops over the
// 16-column group give the full 32-lane total in every lane.
__device__ __forceinline__ float wave_sum_wmma(float v) {
  v2f a; a[0] = 1.0f; a[1] = 1.0f;     // A: 16x4 all-ones
  v2f b; b[0] = v;    b[1] = 0.0f;     // B: per-lane partial in one slot
  v8f c = {};
  v8f d = __builtin_amdgcn_wmma_f32_16x16x4_f32(
      /*neg_a=*/false, a, /*neg_b=*/false, b,
      /*c_mod=*/(short)0, c, /*reuse_a=*/false, /*reuse_b=*/false);
  float s = d[0];
  s += __shfl_xor(s, 1, 32);
  s += __shfl_xor(s, 2, 32);
  s += __shfl_xor(s, 4, 32);
  s += __shfl_xor(s, 8, 32);
  return s;
}

// Kernel 1: one workgroup (32 full waves -> EXEC all-ones for WMMA).
// Materializes weight[S,J] into workspace and reduces both scalars.
__global__ __launch_bounds__(1024)
void bel_prep_reduce(const float* __restrict__ mu,
                     const float* __restrict__ rho,
                     const float* __restrict__ eps,
                     float* __restrict__ w_out,
                     float* __restrict__ scalars /* [0]=log_prior, [1]=log_var_post */) {
  __shared__ float red0[32];
  __shared__ float red1[32];

  const int tid = threadIdx.x;
  float lvp = 0.0f;   // log variational posterior partial
  float lpr = 0.0f;   // log prior partial

  // 16384 elements / 1024 threads = 16 iterations, perfectly uniform.
  for (int i = tid; i < SDIM * JDIM; i += 1024) {
    const int j = i & (JDIM - 1);
    const float m  = mu[j];
    const float sg = log1pf(expf(rho[j]));      // softplus
    const float w  = fmaf(sg, eps[i], m);       // reparameterized sample
    w_out[i] = w;

    // Normal logpdf(w; m, sg)
    const float dv = w - m;
    lvp += -LOG_SQRT_2PI - logf(sg) - (dv * dv) / (2.0f * sg * sg);

    // scale-mixture prior via stable logaddexp
    const float w2  = w * w;
    const float lp1 = -LOG_SQRT_2PI - 0.5f * w2 + LOG_HALF;            // sigma1 = 1
    const float lp2 = -LOG_SQRT_2PI - LOG_SIGMA2 - INV_2SIG2_2 * w2 + LOG_HALF;
    const float mx  = fmaxf(lp1, lp2);
    const float mn  = fminf(lp1, lp2);
    lpr += mx + log1pf(expf(mn - mx));
  }

  // wave32 reduction through the matrix pipe (EXEC is all 1s here)
  const int lane = tid & 31;
  const int wv   = tid >> 5;
  const float s0 = wave_sum_wmma(lvp);
  const float s1 = wave_sum_wmma(lpr);
  if (lane == 0) { red0[wv] = s0; red1[wv] = s1; }
  __syncthreads();

  if (wv == 0) {
    float t0 = red0[lane];
    float t1 = red1[lane];
    for (int msk = 16; msk >= 1; msk >>= 1) {
      t0 += __shfl_xor(t0, msk, 32);
      t1 += __shfl_xor(t1, msk, 32);
    }
    if (lane == 0) {
      scalars[0] = t1;  // log_prior
      scalars[1] = t0;  // log_variational_posterior
    }
  }
}

// Kernel 2: streaming broadcast-multiply.
// Each thread owns 4 j-columns (one float4), holds the 4 per-sample weight
// quads in 16 VGPRs, and loops over rows: 1 NT b128 load of x, 4 NT b128
// stores of out. Consecutive lanes -> consecutive 16B -> fully coalesced.
__global__ __launch_bounds__(256)
void bel_mul(const float* __restrict__ x,
             const float* __restrict__ w,
             float* __restrict__ out,
             int rows_per_block) {
  const int j4 = blockIdx.x * 256 + threadIdx.x;       // 0 .. J4-1
  const v4f* __restrict__ xq = (const v4f*)x;
  const v4f* __restrict__ wq = (const v4f*)w;
  v4f*       __restrict__ oq = (v4f*)out;

  v4f wr[SDIM];
#pragma unroll
  for (int s = 0; s < SDIM; ++s) wr[s] = wq[(size_t)s * J4 + j4];

  const int b0 = blockIdx.y * rows_per_block;
#pragma unroll 2
  for (int r = 0; r < rows_per_block; ++r) {
    const int b = b0 + r;
    v4f xv = __builtin_nontemporal_load(&xq[(size_t)b * J4 + j4]);  // single-use stream
#pragma unroll
    for (int s = 0; s < SDIM; ++s) {
      v4f o = xv * wr[s];
      __builtin_nontemporal_store(o, &oq[((size_t)s * BDIM + b) * J4 + j4]);
    }
  }
}

extern "C" void kernel_launch(void* const* d_in, const int* in_sizes, int n_in,
                              void* d_out, int out_size, void* d_ws, size_t ws_size,
                              hipStream_t stream) {
  const float* x   = (const float*)d_in[0];  // [B, J]
  const float* mu  = (const float*)d_in[1];  // [J]
  const float* rho = (const float*)d_in[2];  // [J]
  const float* eps = (const float*)d_in[3];  // [S, J]
  // d_in[4] = n_samples (compile-time SDIM here)

  float* out = (float*)d_out;
  float* w   = (float*)d_ws;                 // weight[S,J] = 64 KB scratch

  const size_t base = (size_t)SDIM * BDIM * JDIM;  // start of the two scalars

  // 1) weights + both scalar reductions (stream-ordered before the multiply)
  bel_prep_reduce<<<1, 1024, 0, stream>>>(mu, rho, eps, w, out + base);

  // 2) 256 MB NT streaming multiply: grid (J4/256, 256), 16 rows per block
  dim3 grid(J4 / 256, 256);
  bel_mul<<<grid, 256, 0, stream>>>(x, w, out, BDIM / 256);
}